// MambaUNet_27728308863671
// MI455X (gfx1250) — compile-verified
//
#include <hip/hip_runtime.h>
#include <hip/hip_bf16.h>

// ---------------------------------------------------------------------------
// MambaUNet on gfx1250 (MI455X). All projections run on the WMMA matrix pipe
// via V_WMMA_F32_16X16X4_F32 (fp32 in / fp32 acc, matches reference dtype).
// Activations stay in token-major [B, L, C] layout throughout.
// ---------------------------------------------------------------------------

typedef float v2f_ __attribute__((ext_vector_type(2)));
typedef float v8f_ __attribute__((ext_vector_type(8)));

#define ACT_NONE      0
#define ACT_SOFTPLUS  1

static inline int cdiv_(int a, int b) { return (a + b - 1) / b; }

__device__ __forceinline__ float sigmoidf_(float x) { return 1.0f / (1.0f + __expf(-x)); }
__device__ __forceinline__ float siluf_(float x)    { return x * sigmoidf_(x); }
__device__ __forceinline__ float softplusf_(float x){ return (x > 20.0f) ? x : log1pf(__expf(x)); }

// ---------------------------------------------------------------------------
// GEMM: C[M,N] = act( A[M,K] (row stride lda) * W[N,K]^T + bias )
// One wave computes a 16x32 tile (two 16x16 WMMA subtiles sharing the A
// fragment); 8 waves (256 thr) per block -> 128x32 per block.
// Software-pipelined K loop: operands for k+4 are in flight while the two
// v_wmma_f32_16x16x4_f32 for k execute, hiding global load latency.
// VGPR layouts per CDNA5 ISA 7.12.2:
//   A 16x4 : lane l holds A[l%16][ (l/16)*2 + v ]        (v = vgpr 0..1)
//   B 4x16 : lane l holds B[ (l/16)*2 + v ][ l%16 ]  = W[l%16][ (l/16)*2+v ]
//   C 16x16: lane l, vgpr r holds C[ (l/16)*8 + r ][ l%16 ]
// ---------------------------------------------------------------------------
__global__ __launch_bounds__(256) void gemm_wmma_f32_kernel(
    const float* __restrict__ A, int lda,
    const float* __restrict__ Wt,       // [N,K] row-major (i.e. B^T)
    const float* __restrict__ bias,     // [N] or nullptr
    float* __restrict__ C,              // [M,N] row-major
    int M, int N, int K, int act)
{
    const int lane = threadIdx.x & 31;
    const int wave = threadIdx.x >> 5;
    const int half = lane >> 4;        // 0: K 0..1 / rows 0..7, 1: K 2..3 / rows 8..15
    const int lrow = lane & 15;

    const int mt  = blockIdx.y * 8 + wave;
    const int nt0 = blockIdx.x * 2;        // two adjacent 16-wide N subtiles
    const int nt1 = nt0 + 1;

    int arow  = mt  * 16 + lrow; if (arow  >= M) arow  = M - 1;   // clamp; stores guarded
    int brow0 = nt0 * 16 + lrow; if (brow0 >= N) brow0 = N - 1;
    int brow1 = nt1 * 16 + lrow; if (brow1 >= N) brow1 = N - 1;

    const float* __restrict__ Ar  = A  + (size_t)arow  * lda + half * 2;
    const float* __restrict__ Wr0 = Wt + (size_t)brow0 * K   + half * 2;
    const float* __restrict__ Wr1 = Wt + (size_t)brow1 * K   + half * 2;

    v8f_ acc0 = {0.f, 0.f, 0.f, 0.f, 0.f, 0.f, 0.f, 0.f};
    v8f_ acc1 = {0.f, 0.f, 0.f, 0.f, 0.f, 0.f, 0.f, 0.f};

    // prologue load (k = 0)
    v2f_ av;  av.x  = Ar[0];  av.y  = Ar[1];
    v2f_ bv0; bv0.x = Wr0[0]; bv0.y = Wr0[1];
    v2f_ bv1; bv1.x = Wr1[0]; bv1.y = Wr1[1];

    for (int k = 4; k < K; k += 4) {
        // issue next-step loads before consuming current fragments
        v2f_ avn;  avn.x  = Ar[k];      avn.y  = Ar[k + 1];
        v2f_ bv0n; bv0n.x = Wr0[k];     bv0n.y = Wr0[k + 1];
        v2f_ bv1n; bv1n.x = Wr1[k];     bv1n.y = Wr1[k + 1];
        acc0 = __builtin_amdgcn_wmma_f32_16x16x4_f32(
            false, av, false, bv0, (short)0, acc0, false, false);
        acc1 = __builtin_amdgcn_wmma_f32_16x16x4_f32(
            false, av, false, bv1, (short)0, acc1, false, false);
        av = avn; bv0 = bv0n; bv1 = bv1n;
    }
    acc0 = __builtin_amdgcn_wmma_f32_16x16x4_f32(
        false, av, false, bv0, (short)0, acc0, false, false);
    acc1 = __builtin_amdgcn_wmma_f32_16x16x4_f32(
        false, av, false, bv1, (short)0, acc1, false, false);

    // epilogue: guarded stores (bias + optional softplus fused)
    {
        const int gn = nt0 * 16 + lrow;
        if (gn < N) {
            float badd = bias ? bias[gn] : 0.0f;
#pragma unroll
            for (int r = 0; r < 8; ++r) {
                int gm = mt * 16 + half * 8 + r;
                if (gm < M) {
                    float v = acc0[r] + badd;
                    if (act == ACT_SOFTPLUS) v = softplusf_(v);
                    C[(size_t)gm * N + gn] = v;
                }
            }
        }
    }
    {
        const int gn = nt1 * 16 + lrow;
        if (gn < N) {
            float badd = bias ? bias[gn] : 0.0f;
#pragma unroll
            for (int r = 0; r < 8; ++r) {
                int gm = mt * 16 + half * 8 + r;
                if (gm < M) {
                    float v = acc1[r] + badd;
                    if (act == ACT_SOFTPLUS) v = softplusf_(v);
                    C[(size_t)gm * N + gn] = v;
                }
            }
        }
    }
}

// ---------------------------------------------------------------------------
// First conv: 7x7, 3 -> 64, pad 3, on 2x3x64x64; writes token layout [2,4096,64]
// ---------------------------------------------------------------------------
__global__ void conv_first_kernel(const float* __restrict__ x,
                                  const float* __restrict__ w,
                                  const float* __restrict__ bias,
                                  float* __restrict__ out_tok)
{
    int idx = blockIdx.x * blockDim.x + threadIdx.x;     // total 2*4096*64
    if (idx >= 2 * 4096 * 64) return;
    int co = idx & 63;
    int hw = (idx >> 6) & 4095;
    int b  = idx >> 18;
    int oy = hw >> 6, ox = hw & 63;
    float acc = bias[co];
    for (int ci = 0; ci < 3; ++ci)
        for (int ky = 0; ky < 7; ++ky) {
            int iy = oy + ky - 3;
            if (iy < 0 || iy >= 64) continue;
            for (int kx = 0; kx < 7; ++kx) {
                int ix = ox + kx - 3;
                if (ix < 0 || ix >= 64) continue;
                acc += x[((b * 3 + ci) * 64 + iy) * 64 + ix] *
                       w[((co * 3 + ci) * 7 + ky) * 7 + kx];
            }
        }
    out_tok[((size_t)b * 4096 + hw) * 64 + co] = acc;
}

// ---------------------------------------------------------------------------
// LayerNorm over last dim, one wave32 per row
// ---------------------------------------------------------------------------
__global__ void layernorm_kernel(const float* __restrict__ x,
                                 const float* __restrict__ s,
                                 const float* __restrict__ b,
                                 float* __restrict__ out,
                                 int rows, int C)
{
    int wid  = blockIdx.x * (blockDim.x >> 5) + (threadIdx.x >> 5);
    int lane = threadIdx.x & 31;
    if (wid >= rows) return;
    const float* xr = x + (size_t)wid * C;
    float sum = 0.f, sumsq = 0.f;
    for (int i = lane; i < C; i += 32) { float v = xr[i]; sum += v; sumsq += v * v; }
#pragma unroll
    for (int off = 16; off > 0; off >>= 1) {
        sum   += __shfl_xor(sum,   off, 32);
        sumsq += __shfl_xor(sumsq, off, 32);
    }
    float m    = sum / (float)C;
    float var  = sumsq / (float)C - m * m;
    float rstd = rsqrtf(var + 1e-5f);
    float* orow = out + (size_t)wid * C;
    for (int i = lane; i < C; i += 32)
        orow[i] = (xr[i] - m) * rstd * s[i] + b[i];
}

// ---------------------------------------------------------------------------
// Depthwise causal conv1d (k=4, left pad 3) + SiLU.
// Reads u = xz[:, 0:di] (row stride 2*di), writes dense u buffer [B*L, di].
// ---------------------------------------------------------------------------
__global__ void dwconv_silu_kernel(const float* __restrict__ xz,
                                   const float* __restrict__ cw,  // [di,4]
                                   const float* __restrict__ cb,  // [di]
                                   float* __restrict__ u,
                                   int L, int di)
{
    int idx = blockIdx.x * blockDim.x + threadIdx.x;   // total 2*L*di
    if (idx >= 2 * L * di) return;
    int c  = idx % di;
    int bl = idx / di;
    int l  = bl % L;
    size_t stride = (size_t)(2 * di);
    float acc = cb[c];
#pragma unroll
    for (int j = 0; j < 4; ++j) {
        int ll = l - 3 + j;
        if (ll >= 0)
            acc += xz[(size_t)(bl - (l - ll)) * stride + c] * cw[c * 4 + j];
    }
    u[(size_t)bl * di + c] = siluf_(acc);
}

// ---------------------------------------------------------------------------
// Selective scan (serial recurrence), one thread per (batch, channel).
// h state (n=16) in registers; fused u*D skip and silu(z) gate.
// ---------------------------------------------------------------------------
__global__ void scan_kernel(const float* __restrict__ u,
                            const float* __restrict__ dt,
                            const float* __restrict__ A_log,  // [di,16]
                            const float* __restrict__ dbc,    // [B*L, dbcN]; Bm at +dr, C at +dr+16
                            const float* __restrict__ Dp,     // [di]
                            const float* __restrict__ xz,     // z at col di, row stride 2*di
                            float* __restrict__ y,            // [B*L, di]
                            int L, int di, int dr, int dbcN)
{
    int idx = blockIdx.x * blockDim.x + threadIdx.x;   // total 2*di
    if (idx >= 2 * di) return;
    int b = idx / di, d = idx % di;

    float a[16], h[16];
#pragma unroll
    for (int j = 0; j < 16; ++j) { a[j] = -__expf(A_log[d * 16 + j]); h[j] = 0.f; }
    float Dv = Dp[d];

    for (int l = 0; l < L; ++l) {
        size_t row = (size_t)b * L + l;
        float uv  = u[row * di + d];
        float dtv = dt[row * di + d];
        const float* bc = dbc + row * dbcN + dr;
        float du  = dtv * uv;
        float acc = 0.f;
#pragma unroll
        for (int j = 0; j < 16; ++j) {
            float dA = __expf(dtv * a[j]);
            h[j] = dA * h[j] + du * bc[j];
            acc += h[j] * bc[16 + j];
        }
        float zv = xz[row * (size_t)(2 * di) + di + d];
        y[row * di + d] = (acc + uv * Dv) * siluf_(zv);
    }
}

// ---------------------------------------------------------------------------
// Elementwise residual add
// ---------------------------------------------------------------------------
__global__ void add_kernel(const float* __restrict__ a, const float* __restrict__ b,
                           float* __restrict__ o, int n)
{
    int i = blockIdx.x * blockDim.x + threadIdx.x;
    if (i < n) o[i] = a[i] + b[i];
}

// ---------------------------------------------------------------------------
// Patch-merge gather: [B,H*W,C] -> [B,(H/2)*(W/2),4C], quadrant order x0..x3
// ---------------------------------------------------------------------------
__global__ void merge_gather_kernel(const float* __restrict__ t, float* __restrict__ out,
                                    int H, int W, int C)
{
    int H2 = H >> 1, W2 = W >> 1;
    int total = 2 * H2 * W2 * 4 * C;
    int idx = blockIdx.x * blockDim.x + threadIdx.x;
    if (idx >= total) return;
    int c  = idx % C;
    int q  = (idx / C) & 3;
    int l2 = (idx / (4 * C)) % (H2 * W2);
    int b  = idx / (4 * C * H2 * W2);
    int h2 = l2 / W2, w2 = l2 % W2;
    int dy = q & 1, dx = q >> 1;     // q: 0->(0,0) 1->(1,0) 2->(0,1) 3->(1,1)
    out[idx] = t[((size_t)b * H * W + (size_t)(2 * h2 + dy) * W + (2 * w2 + dx)) * C + c];
}

// ---------------------------------------------------------------------------
// Patch-expand pixel shuffle + skip add:
//  t2:[B,H*W,4*Co] -> out:[B,(2H)*(2W),Co];  out = shuffle(t2) + skip
// ---------------------------------------------------------------------------
__global__ void expand_scatter_kernel(const float* __restrict__ t2,
                                      const float* __restrict__ skip,
                                      float* __restrict__ out,
                                      int H, int W, int Co)
{
    int H2 = H * 2, W2 = W * 2;
    int total = 2 * H2 * W2 * Co;
    int idx = blockIdx.x * blockDim.x + threadIdx.x;
    if (idx >= total) return;
    int co = idx % Co;
    int l  = (idx / Co) % (H2 * W2);
    int b  = idx / (Co * H2 * W2);
    int y = l / W2, x = l % W2;
    int h = y >> 1, r = y & 1, w = x >> 1, s = x & 1;
    float v = t2[((size_t)b * H * W + (size_t)h * W + w) * (4 * Co) + (size_t)(r * 2 + s) * Co + co];
    out[idx] = v + skip[idx];
}

// ---------------------------------------------------------------------------
// Final 1x1 conv, 64 -> 1: one wave32 per output pixel
// ---------------------------------------------------------------------------
__global__ void final_conv_kernel(const float* __restrict__ t,
                                  const float* __restrict__ w,
                                  const float* __restrict__ bias,
                                  float* __restrict__ out, int n, int C)
{
    int wid  = blockIdx.x * (blockDim.x >> 5) + (threadIdx.x >> 5);
    int lane = threadIdx.x & 31;
    if (wid >= n) return;
    float acc = 0.f;
    for (int i = lane; i < C; i += 32) acc += t[(size_t)wid * C + i] * w[i];
#pragma unroll
    for (int off = 16; off > 0; off >>= 1) acc += __shfl_xor(acc, off, 32);
    if (lane == 0) out[wid] = acc + bias[0];
}

// ---------------------------------------------------------------------------
// Host orchestration
// ---------------------------------------------------------------------------
struct LNP  { const float* s; const float* b; };
struct MP   { const float *in_w, *conv_w, *conv_b, *x_w, *dt_w, *dt_b, *A_log, *D, *out_w; };
struct VSSP { MP m; LNP mnorm, norm; };

extern "C" void kernel_launch(void* const* d_in, const int* in_sizes, int n_in,
                              void* d_out, int out_size, void* d_ws, size_t ws_size,
                              hipStream_t stream)
{
    (void)in_sizes; (void)n_in; (void)out_size; (void)ws_size;
    static const int DIMSv[4] = {64, 128, 256, 512};

    // ---- parse inputs in setup_inputs() flatten order -----------------------
    int ii = 0;
    auto next = [&]() -> const float* { return (const float*)d_in[ii++]; };
    auto read_ln = [&]() { LNP l; l.s = next(); l.b = next(); return l; };
    auto read_mamba = [&]() {
        MP m;
        m.in_w = next(); m.conv_w = next(); m.conv_b = next();
        m.x_w = next();  m.dt_w = next();   m.dt_b = next();
        m.A_log = next(); m.D = next();     m.out_w = next();
        return m;
    };
    auto read_vss = [&]() { VSSP v; v.m = read_mamba(); v.mnorm = read_ln(); v.norm = read_ln(); return v; };

    const float* x    = next();
    const float* cf_w = next();
    const float* cf_b = next();

    VSSP enc_v[3][2]; LNP enc_mn[3]; const float* enc_mw[3];
    for (int i = 0; i < 3; ++i) {
        enc_v[i][0] = read_vss();
        enc_v[i][1] = read_vss();
        enc_mn[i]   = read_ln();
        enc_mw[i]   = next();
    }
    VSSP bott[2]; bott[0] = read_vss(); bott[1] = read_vss();
    LNP dec_en[3]; const float* dec_ew[3]; VSSP dec_v[3][2];
    for (int j = 0; j < 3; ++j) {
        dec_en[j]   = read_ln();
        dec_ew[j]   = next();
        dec_v[j][0] = read_vss();
        dec_v[j][1] = read_vss();
    }
    const float* fin_w = next();
    const float* fin_b = next();

    // ---- workspace carve (floats) ------------------------------------------
    float* Wsp = (float*)d_ws;
    size_t off = 0;
    auto arena = [&](size_t nf) { float* p = Wsp + off; off += nf; return p; };
    float* tb    = arena(524288);   // main token buffer [B, L, C]
    float* feat0 = arena(524288);
    float* feat1 = arena(262144);
    float* feat2 = arena(131072);
    float* xzb   = arena(2097152);  // in_proj output [B*L, 2*di]
    float* ub    = arena(1048576);  // conv+silu u, and LN temp for expand
    float* dbcb  = arena(524288);   // x_proj output [B*L, dr+32]
    float* dtb   = arena(1048576);  // softplus(dt)
    float* ymb   = arena(1048576);  // scan output
    float* mob   = arena(524288);   // mamba out / merge concat / expand t2
    float* feats[3] = {feat0, feat1, feat2};

    // ---- launch helpers -----------------------------------------------------
    auto gemm = [&](const float* A, int lda, const float* Wt, const float* bias,
                    float* C, int M, int N, int K, int act) {
        dim3 grid(cdiv_(N, 32), cdiv_(M, 128));
        gemm_wmma_f32_kernel<<<grid, dim3(256), 0, stream>>>(A, lda, Wt, bias, C, M, N, K, act);
    };
    auto lnorm = [&](const float* xi, LNP p, float* o, int rows, int C) {
        layernorm_kernel<<<cdiv_(rows, 8), 256, 0, stream>>>(xi, p.s, p.b, o, rows, C);
    };
    auto addk = [&](const float* a, const float* b, float* o, int n) {
        add_kernel<<<cdiv_(n, 256), 256, 0, stream>>>(a, b, o, n);
    };
    auto mamba_run = [&](const float* tin, float* out, const MP& m, int d, int L) {
        int di = 2 * d, dr = (d + 15) / 16, dbcN = dr + 32, M = 2 * L;
        gemm(tin, d, m.in_w, nullptr, xzb, M, 2 * di, d, ACT_NONE);
        dwconv_silu_kernel<<<cdiv_(M * di, 256), 256, 0, stream>>>(xzb, m.conv_w, m.conv_b, ub, L, di);
        gemm(ub, di, m.x_w, nullptr, dbcb, M, dbcN, di, ACT_NONE);
        gemm(dbcb, dbcN, m.dt_w, m.dt_b, dtb, M, di, dr, ACT_SOFTPLUS);
        scan_kernel<<<cdiv_(2 * di, 128), 128, 0, stream>>>(ub, dtb, m.A_log, dbcb, m.D, xzb, ymb, L, di, dr, dbcN);
        gemm(ymb, di, m.out_w, nullptr, out, M, d, di, ACT_NONE);
    };
    auto vss_run = [&](float* t, const VSSP& v, int d, int L) {
        mamba_run(t, mob, v.m, d, L);
        lnorm(mob, v.mnorm, mob, 2 * L, d);
        addk(t, mob, t, 2 * L * d);
        lnorm(t, v.norm, t, 2 * L, d);
    };

    // ---- forward pass -------------------------------------------------------
    conv_first_kernel<<<cdiv_(2 * 4096 * 64, 256), 256, 0, stream>>>(x, cf_w, cf_b, tb);

    int H = 64, Wd = 64;
    for (int i = 0; i < 3; ++i) {
        int d = DIMSv[i], L = H * Wd;
        hipMemcpyAsync(feats[i], tb, sizeof(float) * (size_t)2 * L * d,
                       hipMemcpyDeviceToDevice, stream);
        vss_run(tb, enc_v[i][0], d, L);
        vss_run(tb, enc_v[i][1], d, L);
        int H2 = H / 2, W2 = Wd / 2, L2 = H2 * W2;
        merge_gather_kernel<<<cdiv_(2 * L2 * 4 * d, 256), 256, 0, stream>>>(tb, mob, H, Wd, d);
        lnorm(mob, enc_mn[i], mob, 2 * L2, 4 * d);
        gemm(mob, 4 * d, enc_mw[i], nullptr, tb, 2 * L2, DIMSv[i + 1], 4 * d, ACT_NONE);
        H = H2; Wd = W2;
    }

    vss_run(tb, bott[0], 512, H * Wd);
    vss_run(tb, bott[1], 512, H * Wd);

    for (int j = 0; j < 3; ++j) {
        int ci = 3 - j, co = ci - 1;
        int din = DIMSv[ci], dout = DIMSv[co], L = H * Wd;
        lnorm(tb, dec_en[j], ub, 2 * L, din);
        gemm(ub, din, dec_ew[j], nullptr, mob, 2 * L, 4 * dout, din, ACT_NONE);
        int H2 = H * 2, W2 = Wd * 2;
        expand_scatter_kernel<<<cdiv_(2 * H2 * W2 * dout, 256), 256, 0, stream>>>(mob, feats[co], tb, H, Wd, dout);
        H = H2; Wd = W2;
        vss_run(tb, dec_v[j][0], dout, H * Wd);
        vss_run(tb, dec_v[j][1], dout, H * Wd);
    }

    final_conv_kernel<<<cdiv_(2 * 4096, 8), 256, 0, stream>>>(tb, fin_w, fin_b, (float*)d_out, 2 * 4096, 64);
}